// SelectiveBoundaryRefiner_34213709480603
// MI455X (gfx1250) — compile-verified
//
#include <hip/hip_runtime.h>
#include <hip/hip_bf16.h>
#include <math.h>

typedef __attribute__((ext_vector_type(2))) float v2f;
typedef __attribute__((ext_vector_type(8))) float v8f;

#define Bb 512
#define Nn 256
#define Ee 1024
#define Dd 256
#define Hh 256
#define Cc 8
#define Uu 64
#define XF 10
#define NEGV (-1.0e9f)

static __device__ __forceinline__ int iclamp(int v, int lo, int hi) {
    return v < lo ? lo : (v > hi ? hi : v);
}

// ---------------------------------------------------------------------------
// Fp32 GEMM via V_WMMA_F32_16X16X4_F32.  C[M,N] = act(A[M,K] @ B[K,N] + bias)
// One wave -> one 16x16 tile.  Block = (32,4) waves -> 64 rows per block.
// A-frag (16x4):  lane<16 holds M=lane,K={0,1}; lane>=16 holds M=lane-16,K={2,3}
// B-frag (4x16):  VGPR v, lanes0-15: K=v, N=lane; lanes16-31: K=v+2, N=lane-16
// C/D (16x16):    VGPR v, lanes0-15: M=v; lanes16-31: M=v+8; N=lane&15
// ---------------------------------------------------------------------------
__global__ __launch_bounds__(128)
void gemm_f32_wmma(const float* __restrict__ A, const float* __restrict__ Bm,
                   const float* __restrict__ bias, float* __restrict__ C,
                   int M, int N, int K, int relu)
{
    const int lane = threadIdx.x;        // 0..31
    const int col  = lane & 15;
    const int half = lane >> 4;          // 0 or 1
    const int n0   = blockIdx.x * 16;
    const int m0   = (blockIdx.y * 4 + threadIdx.y) * 16;
    if (m0 >= M || n0 >= N) return;      // uniform per wave -> EXEC all ones inside

    v8f acc = {0.f, 0.f, 0.f, 0.f, 0.f, 0.f, 0.f, 0.f};
    const float* __restrict__ arow = A + (size_t)(m0 + col) * K;

    for (int k = 0; k < K; k += 4) {
        v2f a;
        a.x = arow[k + 2 * half + 0];
        a.y = arow[k + 2 * half + 1];
        v2f b;
        b.x = Bm[(size_t)(k + 2 * half + 0) * N + n0 + col];
        b.y = Bm[(size_t)(k + 2 * half + 1) * N + n0 + col];
        acc = __builtin_amdgcn_wmma_f32_16x16x4_f32(
            /*neg_a=*/false, a, /*neg_b=*/false, b,
            /*c_mod=*/(short)0, acc, /*reuse_a=*/false, /*reuse_b=*/false);
    }

    const float bv = bias[n0 + col];
#pragma unroll
    for (int v = 0; v < 8; ++v) {
        float val = acc[v] + bv;
        if (relu) val = fmaxf(val, 0.f);
        C[(size_t)(m0 + v + 8 * half) * N + n0 + col] = val;
    }
}

// ---------------------------------------------------------------------------
// Gate: learned_gate = sigmoid(h . Wg + bg); refine_gate = max(lm, hr, lg>0.5)
// ---------------------------------------------------------------------------
__global__ __launch_bounds__(256)
void gate_kernel(const float* __restrict__ h, const float* __restrict__ Wg,
                 const float* __restrict__ bg, const float* __restrict__ margin,
                 const float* __restrict__ brisk, float* __restrict__ gate_ws,
                 float* __restrict__ out_gate)
{
    const int b = blockIdx.x, t = threadIdx.x;
    __shared__ float red[256];
    red[t] = h[(size_t)b * Hh + t] * Wg[t];
    __syncthreads();
    for (int s = 128; s > 0; s >>= 1) {
        if (t < s) red[t] += red[t + s];
        __syncthreads();
    }
    if (t == 0) {
        float lg = 1.f / (1.f + expf(-(red[0] + bg[0])));
        float g = 0.f;
        if (margin[b] < 0.2f) g = 1.f;
        if (brisk[b] > 0.6f)  g = 1.f;
        if (lg > 0.5f)        g = 1.f;
        gate_ws[b]  = g;
        out_gate[b] = g;
    }
}

// ---------------------------------------------------------------------------
// Node delta: hidden = relu(hn_base[b] + sum_j x_j * Wn1[H+j]), delta = hidden.Wn2
// One block per batch row; 8 waves, one wave per node row at a time.
// All loop-invariant weight slices held in per-lane registers (k = lane+32*i):
// zero LDS traffic in the hot loop -> pure v_fmac_f32.
// ---------------------------------------------------------------------------
__global__ __launch_bounds__(256)
void node_delta_kernel(const float* __restrict__ hn_base, const float* __restrict__ Wn1,
                       const float* __restrict__ Wn2, const float* __restrict__ bn2,
                       const float* __restrict__ coords, const float* __restrict__ cs,
                       const float* __restrict__ unc, const int* __restrict__ actions,
                       const int* __restrict__ amask, float* __restrict__ node_delta)
{
    const int b = blockIdx.x, t = threadIdx.x;
    const int lane = t & 31, wave = t >> 5;     // 8 waves of 32

    float base_r[8], w2_r[8], wt_r[XF][8];
#pragma unroll
    for (int i = 0; i < 8; ++i) {
        const int k = lane + 32 * i;
        base_r[i] = hn_base[(size_t)b * Hh + k];
        w2_r[i]   = Wn2[k];
#pragma unroll
        for (int j = 0; j < XF; ++j) wt_r[j][i] = Wn1[(size_t)(Hh + j) * Hh + k];
    }
    const float bn2v = bn2[0];

    for (int n = wave; n < Nn; n += 8) {
        float x[XF];
        const float* cd = coords + ((size_t)b * Nn + n) * Cc;
#pragma unroll
        for (int j = 0; j < 8; ++j) x[j] = cd[j];
        x[8] = cs[(size_t)b * Nn + n];
        int a = iclamp(actions[(size_t)b * Nn + n], 0, Uu - 1);
        x[9] = unc[(size_t)b * Uu + a];
        const float amf = amask[(size_t)b * Nn + n] ? 1.f : 0.f;

        float acc = 0.f;
#pragma unroll
        for (int i = 0; i < 8; ++i) {
            float v = base_r[i];
#pragma unroll
            for (int j = 0; j < XF; ++j) v = fmaf(x[j], wt_r[j][i], v);
            acc = fmaf(fmaxf(v, 0.f), w2_r[i], acc);
        }
        for (int off = 16; off > 0; off >>= 1) acc += __shfl_xor(acc, off, 32);
        if (lane == 0) node_delta[(size_t)b * Nn + n] = (acc + bn2v) * amf;
    }
}

// ---------------------------------------------------------------------------
// Edge delta: same structure, extras = [dst_c - src_c, edge_sens, emf]
// ---------------------------------------------------------------------------
__global__ __launch_bounds__(256)
void edge_delta_kernel(const float* __restrict__ he_base, const float* __restrict__ We1,
                       const float* __restrict__ We2, const float* __restrict__ be2,
                       const float* __restrict__ coords, const float* __restrict__ esens,
                       const int* __restrict__ edges, const int* __restrict__ emask,
                       float* __restrict__ edge_delta)
{
    const int b = blockIdx.x, t = threadIdx.x;
    const int lane = t & 31, wave = t >> 5;

    float base_r[8], w2_r[8], wt_r[XF][8];
#pragma unroll
    for (int i = 0; i < 8; ++i) {
        const int k = lane + 32 * i;
        base_r[i] = he_base[(size_t)b * Hh + k];
        w2_r[i]   = We2[k];
#pragma unroll
        for (int j = 0; j < XF; ++j) wt_r[j][i] = We1[(size_t)(Hh + j) * Hh + k];
    }
    const float be2v = be2[0];

    for (int e = wave; e < Ee; e += 8) {
        const int2 ed = ((const int2*)edges)[(size_t)b * Ee + e];
        const int si = iclamp(ed.x, 0, Nn - 1);
        const int di = iclamp(ed.y, 0, Nn - 1);
        const float* cs_ = coords + ((size_t)b * Nn + si) * Cc;
        const float* cd_ = coords + ((size_t)b * Nn + di) * Cc;
        float x[XF];
#pragma unroll
        for (int j = 0; j < 8; ++j) x[j] = cd_[j] - cs_[j];
        x[8] = esens[(size_t)b * Ee + e];
        const float emf = emask[(size_t)b * Ee + e] ? 1.f : 0.f;
        x[9] = emf;

        float acc = 0.f;
#pragma unroll
        for (int i = 0; i < 8; ++i) {
            float v = base_r[i];
#pragma unroll
            for (int j = 0; j < XF; ++j) v = fmaf(x[j], wt_r[j][i], v);
            acc = fmaf(fmaxf(v, 0.f), w2_r[i], acc);
        }
        for (int off = 16; off > 0; off >>= 1) acc += __shfl_xor(acc, off, 32);
        if (lane == 0) edge_delta[(size_t)b * Ee + e] = (acc + be2v) * emf;
    }
}

// ---------------------------------------------------------------------------
// Node finalize: mean-center refined scores, top-2 margin/index, unc_term
// One block (256 threads == N) per batch row.
// ---------------------------------------------------------------------------
__global__ __launch_bounds__(256)
void node_finalize_kernel(const float* __restrict__ cs, const float* __restrict__ node_delta,
                          const float* __restrict__ gate_ws, const int* __restrict__ amask,
                          const int* __restrict__ actions, const float* __restrict__ unc,
                          float* __restrict__ out_scores, float* __restrict__ out_margin,
                          int* __restrict__ out_topidx, float* __restrict__ unc_term_ws)
{
    const int b = blockIdx.x, t = threadIdx.x;    // t == node index
    __shared__ float s_amf[256], s_sum[256], s_unc[256];
    __shared__ float v1[256], v2[256];
    __shared__ int   i1[256];

    const float amf  = amask[(size_t)b * Nn + t] ? 1.f : 0.f;
    const float gate = gate_ws[b];
    const float rs   = (cs[(size_t)b * Nn + t] +
                        gate * 0.1f * node_delta[(size_t)b * Nn + t]) * amf;
    int a = iclamp(actions[(size_t)b * Nn + t], 0, Uu - 1);
    const float ug = unc[(size_t)b * Uu + a];

    s_amf[t] = amf; s_sum[t] = rs; s_unc[t] = ug * amf;
    __syncthreads();
    for (int s = 128; s > 0; s >>= 1) {
        if (t < s) {
            s_amf[t] += s_amf[t + s];
            s_sum[t] += s_sum[t + s];
            s_unc[t] += s_unc[t + s];
        }
        __syncthreads();
    }
    const float denom = fmaxf(s_amf[0], 1.f);
    const float rs2   = (rs - s_sum[0] / denom) * amf;
    out_scores[(size_t)b * Nn + t] = rs2;

    // stable top-2 over masked scores
    v1[t] = amf > 0.f ? rs2 : NEGV;
    i1[t] = t;
    v2[t] = -3.0e38f;
    __syncthreads();
    for (int s = 128; s > 0; s >>= 1) {
        if (t < s) {
            float a1 = v1[t],     a2 = v2[t];     int ai = i1[t];
            float b1 = v1[t + s], b2 = v2[t + s]; int bi = i1[t + s];
            if (b1 > a1) { v1[t] = b1; i1[t] = bi; v2[t] = fmaxf(a1, b2); }
            else         { v1[t] = a1; i1[t] = ai; v2[t] = fmaxf(b1, a2); }
        }
        __syncthreads();
    }
    if (t == 0) {
        out_margin[b]  = v1[0] - v2[0];
        out_topidx[b]  = i1[0];
        unc_term_ws[b] = s_unc[0] / denom;
    }
}

// ---------------------------------------------------------------------------
// Edge finalize: refined_edge, masked variance, refined_risk
// One block (256 threads) per batch row; 4 edges per thread.
// ---------------------------------------------------------------------------
__global__ __launch_bounds__(256)
void edge_finalize_kernel(const float* __restrict__ esens, const float* __restrict__ edge_delta,
                          const float* __restrict__ gate_ws, const int* __restrict__ emask,
                          const float* __restrict__ out_margin, const float* __restrict__ unc_term_ws,
                          float* __restrict__ out_edge, float* __restrict__ out_risk)
{
    const int b = blockIdx.x, t = threadIdx.x;
    const float gate = gate_ws[b];
    float re[4], em[4];
    float sum = 0.f, cnt = 0.f;
#pragma unroll
    for (int i = 0; i < 4; ++i) {
        const int e = t + i * 256;
        const float emf = emask[(size_t)b * Ee + e] ? 1.f : 0.f;
        const float r = (esens[(size_t)b * Ee + e] +
                         gate * 0.1f * edge_delta[(size_t)b * Ee + e]) * emf;
        out_edge[(size_t)b * Ee + e] = r;
        re[i] = r; em[i] = emf;
        sum += r; cnt += emf;        // r already carries *emf
    }
    __shared__ float ssum[256], scnt[256];
    ssum[t] = sum; scnt[t] = cnt;
    __syncthreads();
    for (int s = 128; s > 0; s >>= 1) {
        if (t < s) { ssum[t] += ssum[t + s]; scnt[t] += scnt[t + s]; }
        __syncthreads();
    }
    const float cntT  = scnt[0];
    const float cnt_c = fmaxf(cntT, 1.f);
    const float mean  = ssum[0] / cnt_c;
    __syncthreads();
    float vp = 0.f;
#pragma unroll
    for (int i = 0; i < 4; ++i) vp += em[i] * (re[i] - mean) * (re[i] - mean);
    ssum[t] = vp;
    __syncthreads();
    for (int s = 128; s > 0; s >>= 1) {
        if (t < s) ssum[t] += ssum[t + s];
        __syncthreads();
    }
    if (t == 0) {
        const float var = ssum[0] / cnt_c;
        const float ev  = (cntT > 1.f) ? var : 0.f;
        const float m   = out_margin[b];
        float risk = 1.f / (1.f + expf(m));          // sigmoid(-m)
        risk += 0.2f * ev + 0.1f * unc_term_ws[b];
        out_risk[b] = fminf(fmaxf(risk, 0.f), 1.f);
    }
}

// ---------------------------------------------------------------------------
extern "C" void kernel_launch(void* const* d_in, const int* in_sizes, int n_in,
                              void* d_out, int out_size, void* d_ws, size_t ws_size,
                              hipStream_t stream) {
    const float* features = (const float*)d_in[0];
    const float* cs       = (const float*)d_in[1];
    const float* margin   = (const float*)d_in[2];
    const float* brisk    = (const float*)d_in[3];
    const float* esens    = (const float*)d_in[4];
    const float* coords   = (const float*)d_in[5];
    const float* unc      = (const float*)d_in[6];
    const float* Wc1 = (const float*)d_in[7];   const float* bc1 = (const float*)d_in[8];
    const float* Wc2 = (const float*)d_in[9];   const float* bc2 = (const float*)d_in[10];
    const float* Wn1 = (const float*)d_in[11];  const float* bn1 = (const float*)d_in[12];
    const float* Wn2 = (const float*)d_in[13];  const float* bn2 = (const float*)d_in[14];
    const float* We1 = (const float*)d_in[15];  const float* be1 = (const float*)d_in[16];
    const float* We2 = (const float*)d_in[17];  const float* be2 = (const float*)d_in[18];
    const float* Wg  = (const float*)d_in[19];  const float* bg  = (const float*)d_in[20];
    const int* amask   = (const int*)d_in[21];
    const int* emask   = (const int*)d_in[22];
    const int* actions = (const int*)d_in[23];
    const int* edges   = (const int*)d_in[24];

    float* out = (float*)d_out;
    float* out_scores = out;                         // B*N
    float* out_margin = out + (size_t)Bb * Nn;       // B
    float* out_edge   = out_margin + Bb;             // B*E
    int*   out_topidx = (int*)(out_edge + (size_t)Bb * Ee);  // B (int32 bits)
    float* out_risk   = (float*)(out_topidx + Bb);   // B
    float* out_gate   = out_risk + Bb;               // B

    float* ws = (float*)d_ws;
    float* h1         = ws;                          // B*H
    float* h          = h1 + (size_t)Bb * Hh;        // B*H
    float* hn_base    = h + (size_t)Bb * Hh;         // B*H
    float* he_base    = hn_base + (size_t)Bb * Hh;   // B*H
    float* gate_ws    = he_base + (size_t)Bb * Hh;   // B
    float* node_delta = gate_ws + Bb;                // B*N
    float* edge_delta = node_delta + (size_t)Bb * Nn;// B*E
    float* unc_term   = edge_delta + (size_t)Bb * Ee;// B

    const dim3 gblk(32, 4);
    const dim3 ggrd(Hh / 16, Bb / 64);   // N-tiles x M-tiles(4 waves each)

    // h1 = relu(features @ Wc1 + bc1); h = relu(h1 @ Wc2 + bc2)
    gemm_f32_wmma<<<ggrd, gblk, 0, stream>>>(features, Wc1, bc1, h1, Bb, Hh, Dd, 1);
    gemm_f32_wmma<<<ggrd, gblk, 0, stream>>>(h1, Wc2, bc2, h, Bb, Hh, Hh, 1);
    // hoisted first-layer contributions of h (rows 0..255 of Wn1 / We1)
    gemm_f32_wmma<<<ggrd, gblk, 0, stream>>>(h, Wn1, bn1, hn_base, Bb, Hh, Hh, 0);
    gemm_f32_wmma<<<ggrd, gblk, 0, stream>>>(h, We1, be1, he_base, Bb, Hh, Hh, 0);

    gate_kernel<<<Bb, 256, 0, stream>>>(h, Wg, bg, margin, brisk, gate_ws, out_gate);

    node_delta_kernel<<<Bb, 256, 0, stream>>>(hn_base, Wn1, Wn2, bn2, coords, cs,
                                              unc, actions, amask, node_delta);
    edge_delta_kernel<<<Bb, 256, 0, stream>>>(he_base, We1, We2, be2, coords, esens,
                                              edges, emask, edge_delta);

    node_finalize_kernel<<<Bb, 256, 0, stream>>>(cs, node_delta, gate_ws, amask, actions,
                                                 unc, out_scores, out_margin, out_topidx,
                                                 unc_term);
    edge_finalize_kernel<<<Bb, 256, 0, stream>>>(esens, edge_delta, gate_ws, emask,
                                                 out_margin, unc_term, out_edge, out_risk);
}